// SelfBallPointQuery_56736517980692
// MI455X (gfx1250) — compile-verified
//
#include <hip/hip_runtime.h>

#define NPTS 2048
#define NBATCH 16
#define RAD2 0.04f
#define MAXS 64

typedef float v2f __attribute__((ext_vector_type(2)));
typedef float v8f __attribute__((ext_vector_type(8)));

// One wave (32 lanes) handles 16 query rows of one batch.
// Distances computed 16x16 per V_WMMA_F32_16X16X4_F32 via
// d2 = |xi|^2 + |xj|^2 - 2 xi.xj  (A=[x,y,z,|x|^2], B=[-2x,-2y,-2z,1], C=|xj|^2).
__global__ __launch_bounds__(32) void SelfBallPointQuery_wmma_kernel(
    const float* __restrict__ pcs, long long* __restrict__ out) {
  const int lane = threadIdx.x;            // 0..31
  const int wid  = blockIdx.x;             // 0 .. NBATCH*(NPTS/16)-1
  const int b    = wid >> 7;               // / (NPTS/16)
  const int i0   = (wid & 127) << 4;       // first query row of this wave

  const float* __restrict__ px = pcs + (size_t)b * 3 * NPTS;
  const float* __restrict__ py = px + NPTS;
  const float* __restrict__ pz = py + NPTS;

  // ---- A matrix: 16x4 f32, per ISA layout (lanes 0-15: K=0,1 ; lanes 16-31: K=2,3)
  const int rowA = i0 + (lane & 15);
  const float xi = px[rowA], yi = py[rowA], zi = pz[rowA];
  const float ni = xi * xi + yi * yi + zi * zi;
  v2f a;
  a[0] = (lane < 16) ? xi : zi;   // K=0 (x) | K=2 (z)
  a[1] = (lane < 16) ? yi : ni;   // K=1 (y) | K=3 (|xi|^2)

  // ---- per-row compaction state: lane m (0..15) owns query row i0+m
  int cnt = 0;
  int firstIdx = i0 + lane;                // self is always within radius
  long long* __restrict__ orow =
      out + ((size_t)b * NPTS + (size_t)(i0 + lane)) * MAXS;

  for (int jt = 0; jt < NPTS / 16; ++jt) {
    const int j0 = jt << 4;
    const int cj = j0 + (lane & 15);
    const float xj = px[cj], yj = py[cj], zj = pz[cj];
    const float nj = xj * xj + yj * yj + zj * zj;

    // B matrix 4x16: VGPR0 lanes0-15 = K=0 row, lanes16-31 = K=2 row; VGPR1 = K=1 | K=3
    v2f bm;
    bm[0] = (lane < 16) ? (-2.0f * xj) : (-2.0f * zj);
    bm[1] = (lane < 16) ? (-2.0f * yj) : 1.0f;

    // C: column-constant |xj|^2 (col index = lane&15 for both lane halves)
    v8f c;
#pragma unroll
    for (int v = 0; v < 8; ++v) c[v] = nj;

    // D[v], lanes0-15 -> row v; lanes16-31 -> row v+8 ; col = lane&15
    v8f d = __builtin_amdgcn_wmma_f32_16x16x4_f32(false, a, false, bm,
                                                  (short)0, c, false, false);

    unsigned bal[8];
#pragma unroll
    for (int v = 0; v < 8; ++v)
      bal[v] = __builtin_amdgcn_ballot_w32(d[v] < RAD2);

    // Route ballot halves: row m<8 -> bal[m] low16, row m>=8 -> bal[m-8] high16.
    const int m = lane & 7;
    unsigned t0 = (m & 1) ? bal[1] : bal[0];
    unsigned t1 = (m & 1) ? bal[3] : bal[2];
    unsigned t2 = (m & 1) ? bal[5] : bal[4];
    unsigned t3 = (m & 1) ? bal[7] : bal[6];
    unsigned u0 = (m & 2) ? t1 : t0;
    unsigned u1 = (m & 2) ? t3 : t2;
    unsigned sel = (m & 4) ? u1 : u0;
    unsigned bits =
        (lane < 16) ? ((lane < 8) ? (sel & 0xffffu) : (sel >> 16)) : 0u;

    // Ordered extraction of set bits -> ascending j, capped at 64.
    while (bits) {
      const int bit = __builtin_ctz(bits);
      bits &= bits - 1u;
      if (cnt < MAXS) {
        const int j = j0 + bit;
        if (cnt == 0) firstIdx = j;
        orow[cnt] = (long long)j;
        ++cnt;
      }
    }

    // Uniform early exit when all 16 owned rows are full.
    unsigned doneMask =
        __builtin_amdgcn_ballot_w32((cnt >= MAXS) || (lane >= 16));
    if (doneMask == 0xffffffffu) break;
  }

  // Pad tail with the first (smallest) neighbor index, matching the reference.
  if (lane < 16) {
    for (int k = cnt; k < MAXS; ++k) orow[k] = (long long)firstIdx;
  }
}

extern "C" void kernel_launch(void* const* d_in, const int* in_sizes, int n_in,
                              void* d_out, int out_size, void* d_ws,
                              size_t ws_size, hipStream_t stream) {
  (void)in_sizes; (void)n_in; (void)d_ws; (void)ws_size; (void)out_size;
  const float* pcs = (const float*)d_in[0];        // (16, 3, 2048) f32
  long long* out = (long long*)d_out;              // (16, 2048, 64) i64
  const int nblocks = NBATCH * (NPTS / 16);        // one wave per 16 query rows
  SelfBallPointQuery_wmma_kernel<<<nblocks, 32, 0, stream>>>(pcs, out);
}